// PropagatorBlock_111669150398
// MI455X (gfx1250) — compile-verified
//
#include <hip/hip_runtime.h>
#include <cstdint>
#include <cstddef>

// Problem sizes (fixed by the reference)
#define BB 512
#define HH 1024
#define KK 256
#define VV 512
#define MH 4096            // MUL * H
#define EPSR 1e-6f
#define INV_SQRT_K 0.0625f // 1/sqrt(256)
#define WRITE_RATE 0.1f
#define FORGET_RATE 0.02f
#define VC 32              // V-chunk per workgroup in the memory-update kernel

typedef __attribute__((ext_vector_type(2))) float v2f;
typedef __attribute__((ext_vector_type(8))) float v8f;

enum { EPI_NONE = 0, EPI_RESID = 1, EPI_SILU = 2, EPI_RESID_BIAS = 3, EPI_TANH = 4 };

__device__ __forceinline__ float sigmoidf_(float x) { return 1.f / (1.f + __expf(-x)); }

// ---------------------------------------------------------------------------
// Tensor Data Mover support (gfx1250). Guarded so the file always compiles;
// fallback is the plain float4 global<->LDS copy.
// ---------------------------------------------------------------------------
#if defined(__has_builtin)
#if __has_builtin(__builtin_amdgcn_tensor_load_to_lds) && \
    __has_builtin(__builtin_amdgcn_tensor_store_from_lds) && \
    __has_builtin(__builtin_amdgcn_s_wait_tensorcnt)
#define HAVE_TDM 1
#endif
#endif
#ifndef HAVE_TDM
#define HAVE_TDM 0
#endif

#if HAVE_TDM
typedef __attribute__((ext_vector_type(4))) unsigned int u32x4;
typedef __attribute__((ext_vector_type(4))) int i32x4;
typedef __attribute__((ext_vector_type(8))) int i32x8;

// Build a 2D D# for a KKxVC f32 tile (contiguous rows of VC floats, row
// stride VV floats) and issue TENSOR_LOAD_TO_LDS / TENSOR_STORE_FROM_LDS.
// D# layout per CDNA5 ISA.
__device__ __forceinline__ void tdm_tile_2d(const void* global_tile,
                                            unsigned lds_off, bool is_store) {
  const unsigned long long ga = (unsigned long long)global_tile;
  u32x4 g0;
  g0.x = 1u;                                                 // count=1, user D#
  g0.y = lds_off;                                            // lds_addr (bytes)
  g0.z = (unsigned)(ga & 0xFFFFFFFFu);                       // global_addr[31:0]
  g0.w = (unsigned)((ga >> 32) & 0x01FFFFFFu) | (2u << 30);  // ga[56:32]|type=2
  i32x8 g1;
  g1[0] = 0x20000;                                      // data_size=4B, mask=0
  g1[1] = (int)(((unsigned)VV & 0xFFFFu) << 16);        // tensor_dim0 lo16 (=V)
  g1[2] = (int)(((unsigned)VV >> 16) |
                (((unsigned)KK & 0xFFFFu) << 16));      // td0 hi16 | td1 lo16
  g1[3] = (int)(((unsigned)KK >> 16) |
                ((unsigned)VC << 16));                  // td1 hi16 | tile_dim0
  g1[4] = KK;                                           // tile_dim1=K, tile_dim2=0
  g1[5] = VV;                                           // tensor_dim0_stride lo32
  g1[6] = 0;                                            // stride hi | td1_stride lo
  g1[7] = 0;
  i32x4 g2 = {0, 0, 0, 0};
  i32x4 g3 = {0, 0, 0, 0};
#if __clang_major__ >= 23
  i32x8 g4 = {0, 0, 0, 0, 0, 0, 0, 0};
  if (is_store)
    __builtin_amdgcn_tensor_store_from_lds(g0, g1, g2, g3, g4, 0);
  else
    __builtin_amdgcn_tensor_load_to_lds(g0, g1, g2, g3, g4, 0);
#else
  if (is_store)
    __builtin_amdgcn_tensor_store_from_lds(g0, g1, g2, g3, 0);
  else
    __builtin_amdgcn_tensor_load_to_lds(g0, g1, g2, g3, 0);
#endif
}
#endif // HAVE_TDM

// ---------------------------------------------------------------------------
// WMMA f32 GEMM: out[M,N] = epilogue(A[M,Kd] @ W[Kd,N])
// One wave per 16x16 tile; V_WMMA_F32_16X16X4_F32 accumulation (full f32
// fidelity; the layer is HBM-bound so f32 matrix math is free at roofline).
// ---------------------------------------------------------------------------
template <int EPI>
__global__ void wmma_gemm_f32_kernel(const float* __restrict__ A,
                                     const float* __restrict__ W,
                                     const float* __restrict__ bias,
                                     const float* __restrict__ gamma,
                                     const float* __restrict__ resid,
                                     float* __restrict__ out,
                                     int M, int Kd, int N) {
  const int lane = threadIdx.x;      // 0..31
  const int half = lane >> 4;        // 0 or 1
  const int l    = lane & 15;
  const int n0 = blockIdx.x * 16;
  const int m0 = blockIdx.y * 16;

  v8f acc = {0.f, 0.f, 0.f, 0.f, 0.f, 0.f, 0.f, 0.f};

  const float* arow = A + (size_t)(m0 + l) * Kd + half * 2; // 8B-aligned (Kd%4==0)
  const float* wcol = W + (size_t)(half * 2) * N + (n0 + l);

  for (int k = 0; k < Kd; k += 4) {
    v2f a = *(const v2f*)(arow + k);          // {A[m][k+2h], A[m][k+2h+1]}
    const float* wp = wcol + (size_t)k * N;
    v2f b;
    b.x = wp[0];                              // W[k+2h][n]
    b.y = wp[N];                              // W[k+2h+1][n]
    acc = __builtin_amdgcn_wmma_f32_16x16x4_f32(
        /*neg_a=*/false, a, /*neg_b=*/false, b,
        /*c_mod=*/(short)0, acc, /*reuse_a=*/false, /*reuse_b=*/false);
  }

  const int n = n0 + l;
#pragma unroll
  for (int i = 0; i < 8; ++i) {
    const int r = m0 + i + half * 8;
    float v = acc[i];
    if constexpr (EPI == EPI_SILU) {
      v += bias[n];
      v = v * sigmoidf_(v);
    } else if constexpr (EPI == EPI_TANH) {
      v = tanhf(v);
    } else if constexpr (EPI == EPI_RESID) {
      v = resid[(size_t)r * N + n] + gamma[n] * v;
    } else if constexpr (EPI == EPI_RESID_BIAS) {
      v = resid[(size_t)r * N + n] + gamma[n] * (v + bias[n]);
    }
    out[(size_t)r * N + n] = v;
  }
}

// ---------------------------------------------------------------------------
// Parameter-free RMS norm over rows of length n, times per-column scale.
// ---------------------------------------------------------------------------
__global__ void rmsnorm_scale_kernel(const float* __restrict__ in,
                                     const float* __restrict__ scale,
                                     float* __restrict__ out, int n) {
  const int b = blockIdx.x, t = threadIdx.x;
  __shared__ float red[256];
  const float* row = in + (size_t)b * n;
  float ss = 0.f;
  for (int i = t; i < n; i += 256) {
    float v = row[i];
    ss += v * v;
  }
  red[t] = ss;
  __syncthreads();
  for (int s = 128; s > 0; s >>= 1) {
    if (t < s) red[t] += red[t + s];
    __syncthreads();
  }
  const float inv = rsqrtf(red[0] / (float)n + EPSR);
  for (int i = t; i < n; i += 256) out[(size_t)b * n + i] = row[i] * inv * scale[i];
}

// In-place RMS norm of [B, n] rows (n == blockDim.x == 256), times `extra`.
__global__ void rmsnorm_key_kernel(float* __restrict__ buf, int n, float extra) {
  const int b = blockIdx.x, t = threadIdx.x;
  __shared__ float red[256];
  const float v = buf[(size_t)b * n + t];
  red[t] = v * v;
  __syncthreads();
  for (int s = 128; s > 0; s >>= 1) {
    if (t < s) red[t] += red[t + s];
    __syncthreads();
  }
  const float inv = rsqrtf(red[0] / (float)n + EPSR) * extra;
  buf[(size_t)b * n + t] = v * inv;
}

// ---------------------------------------------------------------------------
// read_value[b,v] = sum_k memory[b,k,v] * rk[b,k]   (HBM-bound sweep, 268 MB)
// ---------------------------------------------------------------------------
__global__ void read_value_kernel(const float* __restrict__ memory,
                                  const float* __restrict__ rk,
                                  float* __restrict__ rv) {
  const int b = blockIdx.x, t = threadIdx.x;
  __shared__ float rks[KK];
  rks[t] = rk[(size_t)b * KK + t]; // blockDim == 256 == KK
  __syncthreads();
  const float* mb = memory + (size_t)b * KK * VV;
  float a0 = 0.f, a1 = 0.f;
#pragma unroll 4
  for (int k = 0; k < KK; ++k) {
    const float kw = rks[k];
    a0 += mb[(size_t)k * VV + t] * kw;
    a1 += mb[(size_t)k * VV + t + 256] * kw;
  }
  rv[(size_t)b * VV + t] = a0;
  rv[(size_t)b * VV + t + 256] = a1;
}

// Per-batch write/forget gates: sigmoid(w·wg + b) * rate
__global__ void gates_kernel(const float* __restrict__ w,
                             const float* __restrict__ wg, const float* __restrict__ bwg,
                             const float* __restrict__ fg, const float* __restrict__ bfg,
                             float* __restrict__ gates) {
  const int b = blockIdx.x, t = threadIdx.x;
  __shared__ float s1[256], s2[256];
  float a = 0.f, c = 0.f;
  for (int i = t; i < HH; i += 256) {
    const float x = w[(size_t)b * HH + i];
    a += x * wg[i];
    c += x * fg[i];
  }
  s1[t] = a;
  s2[t] = c;
  __syncthreads();
  for (int s = 128; s > 0; s >>= 1) {
    if (t < s) {
      s1[t] += s1[t + s];
      s2[t] += s2[t + s];
    }
    __syncthreads();
  }
  if (t == 0) {
    gates[b * 2 + 0] = WRITE_RATE * sigmoidf_(s1[0] + bwg[0]);
    gates[b * 2 + 1] = FORGET_RATE * sigmoidf_(s2[0] + bfg[0]);
  }
}

// ---------------------------------------------------------------------------
// Fused value_hat + delta-rule memory update. One workgroup per (b, VC-wide
// V chunk): TDM-stage a 256x32 f32 tile (32 KB) into LDS, reduce over K for
// value_hat, form clipped err, rewrite the tile in LDS, then TDM-store it
// back. One global read + one global write per memory element.
// ---------------------------------------------------------------------------
__global__ void memory_update_kernel(const float* __restrict__ memory,
                                     const float* __restrict__ wk,
                                     const float* __restrict__ wv,
                                     const float* __restrict__ gates,
                                     const unsigned char* __restrict__ valid,
                                     float* __restrict__ out_mem) {
  const int b = blockIdx.y;
  const int v0 = blockIdx.x * VC;
  const int t = threadIdx.x; // 256 threads = 8 waves

  __shared__ float tile[KK][VC]; // 32 KB, same contiguous layout TDM produces
  __shared__ float wks[KK];
  __shared__ float part[8][VC];
  __shared__ float errs[VC];

  wks[t] = wk[(size_t)b * KK + t];

  const float* mb = memory + (size_t)b * KK * VV + v0;
#if HAVE_TDM
  if (t == 0) {
    tdm_tile_2d(mb, (unsigned)(size_t)&tile[0][0], /*is_store=*/false);
  }
  __builtin_amdgcn_s_wait_tensorcnt(0); // non-issuing waves: TENSORcnt==0, no-op
  __syncthreads();
#else
#pragma unroll
  for (int j = 0; j < 8; ++j) {
    const int idx = t + 256 * j; // 0..2047 float4 tiles
    const int k = idx >> 3;
    const int q = idx & 7;
    const float4 g = *(const float4*)(mb + (size_t)k * VV + q * 4);
    *(float4*)&tile[k][q * 4] = g;
  }
  __syncthreads();
#endif

  const int vc = t & (VC - 1);
  const int kh = t >> 5; // one wave == one kh segment (conflict-free LDS reads)
  float acc = 0.f;
#pragma unroll 4
  for (int k = kh * 32; k < kh * 32 + 32; ++k) acc += tile[k][vc] * wks[k];
  part[kh][vc] = acc;
  __syncthreads();

  if (t < VC) {
    float vh = 0.f;
#pragma unroll
    for (int j = 0; j < 8; ++j) vh += part[j][t];
    float e = wv[(size_t)b * VV + v0 + t] - vh;
    errs[t] = fminf(1.f, fmaxf(-1.f, e));
  }
  __syncthreads();

  const float eta = gates[b * 2 + 0];
  const float fgt = gates[b * 2 + 1];
  const bool vld = valid[b] != 0;
  const float e = errs[vc];
#pragma unroll 4
  for (int k = kh; k < KK; k += 8) {
    const float old = tile[k][vc];
    float nm = (1.f - fgt) * old + eta * wks[k] * e;
    nm = fminf(10.f, fmaxf(-10.f, nm));
    tile[k][vc] = vld ? nm : old;
  }
  __syncthreads();

  float* ob = out_mem + (size_t)b * KK * VV + v0;
#if HAVE_TDM
  if (t == 0) {
    tdm_tile_2d(ob, (unsigned)(size_t)&tile[0][0], /*is_store=*/true);
  }
  // S_ENDPGM's implicit wait-idle drains TENSORcnt before LDS is released.
#else
#pragma unroll
  for (int j = 0; j < 8; ++j) {
    const int idx = t + 256 * j;
    const int k = idx >> 3;
    const int q = idx & 7;
    *(float4*)(ob + (size_t)k * VV + q * 4) = *(const float4*)&tile[k][q * 4];
  }
#endif
}

// ---------------------------------------------------------------------------
extern "C" void kernel_launch(void* const* d_in, const int* in_sizes, int n_in,
                              void* d_out, int out_size, void* d_ws, size_t ws_size,
                              hipStream_t stream) {
  const float* x             = (const float*)d_in[0];
  const float* memory        = (const float*)d_in[1];
  const unsigned char* valid = (const unsigned char*)d_in[2];
  const float* w_read_key    = (const float*)d_in[3];
  const float* w_write_key   = (const float*)d_in[4];
  const float* w_write_value = (const float*)d_in[5];
  const float* w_read_proj   = (const float*)d_in[6];
  const float* w_write_gate  = (const float*)d_in[7];
  const float* b_write_gate  = (const float*)d_in[8];
  const float* w_forget_gate = (const float*)d_in[9];
  const float* b_forget_gate = (const float*)d_in[10];
  const float* scale1        = (const float*)d_in[11];
  const float* scale2        = (const float*)d_in[12];
  const float* scale3        = (const float*)d_in[13];
  const float* w_fc1         = (const float*)d_in[14];
  const float* b_fc1         = (const float*)d_in[15];
  const float* w_fc2         = (const float*)d_in[16];
  const float* b_fc2         = (const float*)d_in[17];
  const float* gamma1        = (const float*)d_in[18];
  const float* gamma2        = (const float*)d_in[19];

  float* out_x   = (float*)d_out;                    // [B,H]
  float* out_mem = (float*)d_out + (size_t)BB * HH;  // [B,K,V]

  // Workspace carve-up (~20 MB of f32)
  float* ws = (float*)d_ws;
  float* h_buf = ws;  ws += (size_t)BB * HH;
  float* rk    = ws;  ws += (size_t)BB * KK;
  float* rv    = ws;  ws += (size_t)BB * VV;
  float* x1    = ws;  ws += (size_t)BB * HH;
  float* t2    = ws;  ws += (size_t)BB * HH;
  float* u     = ws;  ws += (size_t)BB * MH;
  float* w3    = ws;  ws += (size_t)BB * HH;
  float* wkb   = ws;  ws += (size_t)BB * KK;
  float* wvb   = ws;  ws += (size_t)BB * VV;
  float* gts   = ws;  ws += (size_t)BB * 2;

  const dim3 wave(32);

  // --- memory read path ---
  rmsnorm_scale_kernel<<<BB, 256, 0, stream>>>(x, scale1, h_buf, HH);
  wmma_gemm_f32_kernel<EPI_NONE><<<dim3(KK / 16, BB / 16), wave, 0, stream>>>(
      h_buf, w_read_key, nullptr, nullptr, nullptr, rk, BB, HH, KK);
  rmsnorm_key_kernel<<<BB, KK, 0, stream>>>(rk, KK, INV_SQRT_K);
  read_value_kernel<<<BB, 256, 0, stream>>>(memory, rk, rv);
  wmma_gemm_f32_kernel<EPI_RESID><<<dim3(HH / 16, BB / 16), wave, 0, stream>>>(
      rv, w_read_proj, nullptr, gamma1, x, x1, BB, VV, HH);

  // --- MLP ---
  rmsnorm_scale_kernel<<<BB, 256, 0, stream>>>(x1, scale2, t2, HH);
  wmma_gemm_f32_kernel<EPI_SILU><<<dim3(MH / 16, BB / 16), wave, 0, stream>>>(
      t2, w_fc1, b_fc1, nullptr, nullptr, u, BB, HH, MH);
  wmma_gemm_f32_kernel<EPI_RESID_BIAS><<<dim3(HH / 16, BB / 16), wave, 0, stream>>>(
      u, w_fc2, b_fc2, gamma2, x1, out_x, BB, MH, HH);

  // --- memory write path ---
  rmsnorm_scale_kernel<<<BB, 256, 0, stream>>>(out_x, scale3, w3, HH);
  wmma_gemm_f32_kernel<EPI_NONE><<<dim3(KK / 16, BB / 16), wave, 0, stream>>>(
      w3, w_write_key, nullptr, nullptr, nullptr, wkb, BB, HH, KK);
  rmsnorm_key_kernel<<<BB, KK, 0, stream>>>(wkb, KK, INV_SQRT_K);
  wmma_gemm_f32_kernel<EPI_TANH><<<dim3(VV / 16, BB / 16), wave, 0, stream>>>(
      w3, w_write_value, nullptr, nullptr, nullptr, wvb, BB, HH, VV);
  gates_kernel<<<BB, 256, 0, stream>>>(w3, w_write_gate, b_write_gate,
                                       w_forget_gate, b_forget_gate, gts);
  memory_update_kernel<<<dim3(VV / VC, BB), 256, 0, stream>>>(
      memory, wkb, wvb, gts, valid, out_mem);
}